// EdgeDegreeEmbeddingNetwork_20160576488089
// MI455X (gfx1250) — compile-verified
//
#include <hip/hip_runtime.h>
#include <hip/hip_bf16.h>

typedef float v2f __attribute__((ext_vector_type(2)));
typedef float v8f __attribute__((ext_vector_type(8)));

#define N_NODES 10000
#define N_EDGES 100000
#define OUT_DIM 240
#define NTILES  ((N_EDGES + 15) / 16)   // 6250
#define WAVES   8

// ---------------------------------------------------------------------------
// Kernel 1: fold weights into Q (112x32) and r (112).
//   g[n] = sum_k h[k] * Q[n,k] + r[n]
//   n<64      : Q[n,k] = sum_c s_c * proj_w0[n,c]    * rad_w2[c,k]
//   64<=n<96  : Q[n,k] = sum_c s_c * proj_w1[n-64,c] * rad_w2[64+c,k]
//   96<=n<112 : Q[n,k] = sum_c s_c * proj_w2[n-96,c] * rad_w2[128+c,k]
//   r[n]      : same with rad_offset instead of rad_w2 column
// ---------------------------------------------------------------------------
__global__ void qr_fold_kernel(const float* __restrict__ exp_w,
                               const float* __restrict__ exp_b,
                               const float* __restrict__ rad_w2,
                               const float* __restrict__ rad_offset,
                               const float* __restrict__ pw0,
                               const float* __restrict__ pw1,
                               const float* __restrict__ pw2,
                               float* __restrict__ Q,
                               float* __restrict__ R) {
  __shared__ float s[64];
  if (threadIdx.x < 64) s[threadIdx.x] = exp_w[threadIdx.x] + exp_b[threadIdx.x];
  __syncthreads();
  int idx = blockIdx.x * blockDim.x + threadIdx.x;
  if (idx < 112 * 32) {
    int n = idx >> 5, k = idx & 31;
    const float* pw; int d, off, stride;
    if (n < 64)      { pw = pw0; d = n;      off = 0;   stride = 112; }
    else if (n < 96) { pw = pw1; d = n - 64; off = 64;  stride = 192; }
    else             { pw = pw2; d = n - 96; off = 128; stride = 192; }
    float acc = 0.f;
    for (int c = 0; c < 64; ++c)
      acc += s[c] * pw[d * stride + c] * rad_w2[(off + c) * 32 + k];
    Q[idx] = acc;
  } else if (idx < 112 * 32 + 112) {
    int n = idx - 112 * 32;
    const float* pw; int d, off, stride;
    if (n < 64)      { pw = pw0; d = n;      off = 0;   stride = 112; }
    else if (n < 96) { pw = pw1; d = n - 64; off = 64;  stride = 192; }
    else             { pw = pw2; d = n - 96; off = 128; stride = 192; }
    float acc = 0.f;
    for (int c = 0; c < 64; ++c)
      acc += s[c] * pw[d * stride + c] * rad_offset[off + c];
    R[n] = acc;
  }
}

// ---------------------------------------------------------------------------
// Kernel 2: zero output (we accumulate with atomics).
// ---------------------------------------------------------------------------
__global__ void zero_out_kernel(float* __restrict__ out, int n) {
  int i = blockIdx.x * blockDim.x + threadIdx.x;
  if (i < n) out[i] = 0.f;
}

// ---------------------------------------------------------------------------
// Kernel 3: main edge kernel. One wave32 per 16-edge tile.
//   GEMM A:  h(16x32)  = X(16x32)   @ rad_w1^T     -> 16 v_wmma_f32_16x16x4_f32
//   LN + SiLU rows
//   GEMM B:  g(16x112) = h(16x32)   @ Q^T          -> 56 v_wmma_f32_16x16x4_f32
//   Epilogue: outer products with a0/a1/a2, atomic scatter to node_out.
// ---------------------------------------------------------------------------
__global__ __launch_bounds__(256)
void edge_deg_embed_kernel(const float* __restrict__ edge_attr,     // E x 9
                           const float* __restrict__ edge_scalars,  // E x 32
                           const float* __restrict__ rad_w1,        // 32 x 32
                           const float* __restrict__ lng,           // 32
                           const float* __restrict__ lnb,           // 32
                           const float* __restrict__ pb0,           // 64
                           const int*   __restrict__ edge_dst,      // E
                           const float* __restrict__ Qg,            // 112 x 32
                           const float* __restrict__ Rg,            // 112
                           float* __restrict__ out)                 // N x 240
{
  // Padded LDS (row stride 34 floats: bank-conflict-free, keeps float2 8B-aligned)
  __shared__ float sW1[32 * 34];
  __shared__ float sQ[112 * 34];
  __shared__ float sR[112];
  __shared__ float sPB[64];
  __shared__ float sLG[32];
  __shared__ float sLB[32];
  __shared__ float sH[WAVES][16 * 34];
  __shared__ float sEM[WAVES][16 * 10];   // per edge: dst(bits), a0, a1[3], a2[5]

  const int tid  = threadIdx.x;
  const int wave = tid >> 5;
  const int lane = tid & 31;
  const int nn   = lane & 15;
  const int half = lane >> 4;
  const int kb   = half * 2;

  for (int i = tid; i < 1024; i += 256) sW1[(i >> 5) * 34 + (i & 31)] = rad_w1[i];
  for (int i = tid; i < 3584; i += 256) sQ[(i >> 5) * 34 + (i & 31)] = Qg[i];
  for (int i = tid; i < 112;  i += 256) sR[i]  = Rg[i];
  for (int i = tid; i < 64;   i += 256) sPB[i] = pb0[i];
  for (int i = tid; i < 32;   i += 256) { sLG[i] = lng[i]; sLB[i] = lnb[i]; }
  __syncthreads();

  const float rs112 = 0.09449111825230681f;  // 1/sqrt(112)
  const float rs192 = 0.07216878364870323f;  // 1/sqrt(192)
  const float agg   = 0.25f;                 // 1/sqrt(AVG_AGG)

  for (int tile = blockIdx.x * WAVES + wave; tile < NTILES; tile += gridDim.x * WAVES) {
    const int base = tile * 16;
    float* em = &sEM[wave][0];

    // ---- per-edge metadata into LDS (lanes 0..15) ----
    if (lane < 16) {
      const int e = base + lane;
      float* r = em + lane * 10;
      if (e < N_EDGES) {
        r[0] = __int_as_float(edge_dst[e]);
        const float* ea = edge_attr + (size_t)e * 9;
#pragma unroll
        for (int j = 0; j < 9; ++j) r[1 + j] = ea[j];
      } else {
        r[0] = __int_as_float(-1);
#pragma unroll
        for (int j = 0; j < 9; ++j) r[1 + j] = 0.f;
      }
    }

    // ---- GEMM A: h = X @ rad_w1^T  (fp32 WMMA, K=32 in 8 steps, N-tiles=2) ----
    const int erow  = base + nn;
    const bool vld  = erow < N_EDGES;
    const float* xr = edge_scalars + (size_t)erow * 32;

    v8f hAcc0 = {0.f, 0.f, 0.f, 0.f, 0.f, 0.f, 0.f, 0.f};
    v8f hAcc1 = {0.f, 0.f, 0.f, 0.f, 0.f, 0.f, 0.f, 0.f};
#pragma unroll
    for (int kk = 0; kk < 8; ++kk) {
      const int k = kk * 4 + kb;
      v2f av = {0.f, 0.f};
      if (vld) { av[0] = xr[k]; av[1] = xr[k + 1]; }           // A: M=edge row, K
      v2f b0; b0[0] = sW1[nn * 34 + k];        b0[1] = sW1[nn * 34 + k + 1];
      v2f b1; b1[0] = sW1[(16 + nn) * 34 + k]; b1[1] = sW1[(16 + nn) * 34 + k + 1];
      hAcc0 = __builtin_amdgcn_wmma_f32_16x16x4_f32(false, av, false, b0, (short)0, hAcc0, false, false);
      hAcc1 = __builtin_amdgcn_wmma_f32_16x16x4_f32(false, av, false, b1, (short)0, hAcc1, false, false);
    }

    // ---- spill h (C layout -> row-major LDS) ----
    float* ht = &sH[wave][0];
#pragma unroll
    for (int r = 0; r < 8; ++r) {
      ht[(r + 8 * half) * 34 + nn]      = hAcc0[r];
      ht[(r + 8 * half) * 34 + 16 + nn] = hAcc1[r];
    }
    asm volatile("s_wait_dscnt 0" ::: "memory");

    // ---- LayerNorm + SiLU per edge row (lanes 0..15, one row each) ----
    if (lane < 16) {
      float* hr = ht + lane * 34;
      float s1 = 0.f, s2 = 0.f;
#pragma unroll
      for (int k = 0; k < 32; ++k) { const float v = hr[k]; s1 += v; s2 += v * v; }
      const float mu  = s1 * 0.03125f;
      const float var = s2 * 0.03125f - mu * mu;
      const float rsd = rsqrtf(var + 1e-5f);
#pragma unroll
      for (int k = 0; k < 32; ++k) {
        float v = (hr[k] - mu) * rsd * sLG[k] + sLB[k];
        v = v * (1.f / (1.f + __expf(-v)));                    // SiLU
        hr[k] = v;
      }
    }
    asm volatile("s_wait_dscnt 0" ::: "memory");

    // ---- GEMM B: g = h @ Q^T  (N=112 -> 7 tiles, K=32 -> 8 steps) ----
    v8f g[7];
#pragma unroll
    for (int t = 0; t < 7; ++t) g[t] = (v8f){0.f, 0.f, 0.f, 0.f, 0.f, 0.f, 0.f, 0.f};

    const float* ar = ht + nn * 34;
#pragma unroll
    for (int kk = 0; kk < 8; ++kk) {
      const int k = kk * 4 + kb;
      v2f av; av[0] = ar[k]; av[1] = ar[k + 1];
#pragma unroll
      for (int t = 0; t < 7; ++t) {
        const float* qr = sQ + (16 * t + nn) * 34 + k;
        v2f bv; bv[0] = qr[0]; bv[1] = qr[1];
        g[t] = __builtin_amdgcn_wmma_f32_16x16x4_f32(false, av, false, bv, (short)0, g[t], false, false);
      }
    }

    // ---- epilogue: outer products + atomic scatter ----
#pragma unroll
    for (int t = 0; t < 7; ++t) {
      const int n  = 16 * t + nn;
      const float rb = sR[n];
#pragma unroll
      for (int r = 0; r < 8; ++r) {
        const int m = r + 8 * half;
        const float* e = em + m * 10;
        const int dst = __float_as_int(e[0]);
        if (dst < 0) continue;
        const float gv = g[t][r] + rb;
        float* orow = out + (size_t)dst * OUT_DIM;
        if (t < 4) {                       // n in [0,64): scalar (l=0) block
          atomicAdd(orow + n, (gv * e[1] * rs112 + sPB[n]) * agg);
        } else if (t < 6) {                // n in [64,96): l=1 block, 3 components
          const int d = n - 64;
          const float bse = gv * rs192 * agg;
          atomicAdd(orow + 64 + d * 3 + 0, bse * e[2]);
          atomicAdd(orow + 64 + d * 3 + 1, bse * e[3]);
          atomicAdd(orow + 64 + d * 3 + 2, bse * e[4]);
        } else {                           // n in [96,112): l=2 block, 5 components
          const int d = n - 96;
          const float bse = gv * rs192 * agg;
#pragma unroll
          for (int kc = 0; kc < 5; ++kc)
            atomicAdd(orow + 160 + d * 5 + kc, bse * e[5 + kc]);
        }
      }
    }
  }
}

// ---------------------------------------------------------------------------
extern "C" void kernel_launch(void* const* d_in, const int* in_sizes, int n_in,
                              void* d_out, int out_size, void* d_ws, size_t ws_size,
                              hipStream_t stream) {
  const float* edge_attr    = (const float*)d_in[1];
  const float* edge_scalars = (const float*)d_in[2];
  const float* exp_w        = (const float*)d_in[3];
  const float* exp_b        = (const float*)d_in[4];
  const float* rad_w1       = (const float*)d_in[5];
  const float* rad_ln_g     = (const float*)d_in[6];
  const float* rad_ln_b     = (const float*)d_in[7];
  const float* rad_w2       = (const float*)d_in[8];
  const float* rad_offset   = (const float*)d_in[9];
  const float* proj_w0      = (const float*)d_in[10];
  const float* proj_b0      = (const float*)d_in[11];
  const float* proj_w1      = (const float*)d_in[12];
  const float* proj_w2      = (const float*)d_in[13];
  const int*   edge_dst     = (const int*)d_in[15];
  float*       out          = (float*)d_out;

  float* Q = (float*)d_ws;          // 112*32 floats
  float* R = Q + 112 * 32;          // 112 floats

  qr_fold_kernel<<<29, 128, 0, stream>>>(exp_w, exp_b, rad_w2, rad_offset,
                                         proj_w0, proj_w1, proj_w2, Q, R);

  const int n_out = N_NODES * OUT_DIM;
  zero_out_kernel<<<(n_out + 511) / 512, 512, 0, stream>>>(out, n_out);

  const int blocks = (NTILES + WAVES - 1) / WAVES;   // 782
  edge_deg_embed_kernel<<<blocks, 256, 0, stream>>>(edge_attr, edge_scalars,
                                                    rad_w1, rad_ln_g, rad_ln_b,
                                                    proj_b0, edge_dst, Q, R, out);
}